// AdaptiveThresholding_25125558681616
// MI455X (gfx1250) — compile-verified
//
#include <hip/hip_runtime.h>

// Adaptive mean thresholding (cv2 ADAPTIVE_THRESH_MEAN_C + THRESH_BINARY_INV)
// 128 x 512 x 512 f32. ~0.5 FLOP/byte -> HBM-bound (~12 us floor at 23.3 TB/s).
// Optimize the HBM->LDS path with CDNA5 async global->LDS DMA (b128 for
// interior tiles, clamped b32 for x-edge tiles), tracked by ASYNCcnt.

#define IMG_H 512
#define IMG_W 512
#define TILE 64
#define KS 11
#define P 5                        // halo = KS/2
#define IN_DIM (TILE + 2 * P)      // 74 rows staged
#define IN_W 80                    // staged cols: x0-8 .. x0+71 (16B aligned)
#define XOFF 8                     // staged col 0 <-> image col x0-8
#define IN_STRIDE 84               // multiple of 4 floats -> 16B-aligned rows
#define HS_STRIDE 65               // gcd(65%64,64)=1 -> conflict-free
#define NTHREADS 256

__device__ __forceinline__ unsigned lds_byte_off(const void* p) {
    // Low 32 bits of a generic pointer to __shared__ = wave-relative LDS byte
    // offset (AS3 value) = what the async-to-LDS VDST operand wants.
    return (unsigned)(uintptr_t)p;
}

__device__ __forceinline__ void async_b32_to_lds(const float* gbase,
                                                 unsigned goff_bytes,
                                                 void* lds_dst) {
    unsigned loff = lds_byte_off(lds_dst);
    asm volatile("global_load_async_to_lds_b32 %0, %1, %2"
                 :: "v"(loff), "v"(goff_bytes), "s"(gbase)
                 : "memory");
}

__device__ __forceinline__ void async_b128_to_lds(const float* gbase,
                                                  unsigned goff_bytes,
                                                  void* lds_dst) {
    unsigned loff = lds_byte_off(lds_dst);
    asm volatile("global_load_async_to_lds_b128 %0, %1, %2"
                 :: "v"(loff), "v"(goff_bytes), "s"(gbase)
                 : "memory");
}

__device__ __forceinline__ void wait_async_done() {
#if __has_builtin(__builtin_amdgcn_s_wait_asynccnt)
    __builtin_amdgcn_s_wait_asynccnt(0);
#else
    asm volatile("s_wait_asynccnt 0x0" ::: "memory");
#endif
}

__device__ __forceinline__ int clampi(int v, int lo, int hi) {
    return v < lo ? lo : (v > hi ? hi : v);
}

__global__ __launch_bounds__(NTHREADS)
void AdaptiveThresholding_25125558681616_kernel(const float* __restrict__ img,
                                                float* __restrict__ out) {
    __shared__ __align__(16) float in_tile[IN_DIM * IN_STRIDE]; // 24.9 KB
    __shared__ float hsum[IN_DIM * HS_STRIDE];                  // 19.2 KB

    const int t  = threadIdx.x;
    const int x0 = blockIdx.x * TILE;
    const int y0 = blockIdx.y * TILE;
    const int b  = blockIdx.z;

    const float* plane  = img + (size_t)b * (IMG_H * IMG_W);
    float*       oplane = out + (size_t)b * (IMG_H * IMG_W);

    // ---- Phase 1: async DMA the staged 74 x 80 window into LDS.
    // Staged col c <-> image col (x0 - 8 + c); y rows clamped (replicate).
    const bool interior_x = (x0 >= XOFF) && (x0 - XOFF + IN_W <= IMG_W);
    if (interior_x) {
        // 74 rows x 20 b128 transfers; global & LDS addresses 16B aligned.
        for (int i = t; i < IN_DIM * (IN_W / 4); i += NTHREADS) {
            int iy = i / (IN_W / 4);
            int j  = i - iy * (IN_W / 4);
            int gy = clampi(y0 - P + iy, 0, IMG_H - 1);
            unsigned goff = (unsigned)((gy * IMG_W + (x0 - XOFF) + 4 * j) * 4);
            async_b128_to_lds(plane, goff, &in_tile[iy * IN_STRIDE + 4 * j]);
        }
    } else {
        // x-edge tiles: per-element clamped b32 path, same LDS layout.
        for (int i = t; i < IN_DIM * IN_W; i += NTHREADS) {
            int iy = i / IN_W;
            int ix = i - iy * IN_W;
            int gy = clampi(y0 - P + iy, 0, IMG_H - 1);
            int gx = clampi(x0 - XOFF + ix, 0, IMG_W - 1);
            unsigned goff = (unsigned)((gy * IMG_W + gx) * 4);
            async_b32_to_lds(plane, goff, &in_tile[iy * IN_STRIDE + ix]);
        }
    }
    wait_async_done();     // this wave's async DMAs landed in LDS
    __syncthreads();       // all waves' DMAs landed

    // ---- Phase 2: horizontal sliding 11-sum. 2 threads per row, running sum.
    // Output col c window = staged cols c+3 .. c+13.
    if (t < 2 * IN_DIM) {
        int r    = t % IN_DIM;      // adjacent lanes -> adjacent rows
        int half = t / IN_DIM;      // 0 or 1
        int c0   = half * (TILE / 2);
        const float* row  = &in_tile[r * IN_STRIDE];
        float*       hrow = &hsum[r * HS_STRIDE];
        float s = 0.0f;
#pragma unroll
        for (int j = 0; j < KS; ++j) s += row[c0 + 3 + j];
        hrow[c0] = s;
        for (int c = c0 + 1; c < c0 + (TILE / 2); ++c) {
            s += row[c + 13] - row[c + 2];
            hrow[c] = s;
        }
    }
    __syncthreads();

    // ---- Phase 3: vertical sliding 11-sum per column, threshold, store.
    {
        const int ox  = t & (TILE - 1);   // 0..63
        const int g   = t >> 6;           // 0..3 row groups of 16
        const int oy0 = g * 16;
        const float inv_area = 1.0f / (float)(KS * KS);

        float s = 0.0f;
#pragma unroll
        for (int j = 0; j < KS; ++j) s += hsum[(oy0 + j) * HS_STRIDE + ox];

#pragma unroll
        for (int i = 0; i < 16; ++i) {
            int oy = oy0 + i;
            if (i > 0)
                s += hsum[(oy + KS - 1) * HS_STRIDE + ox]
                   - hsum[(oy - 1) * HS_STRIDE + ox];
            float mean = s * inv_area;
            float thr  = mean - 2.0f;
            float src  = in_tile[(oy + P) * IN_STRIDE + (ox + XOFF)];
            oplane[(size_t)(y0 + oy) * IMG_W + (x0 + ox)] =
                (src > thr) ? 0.0f : 255.0f;
        }
    }
}

extern "C" void kernel_launch(void* const* d_in, const int* in_sizes, int n_in,
                              void* d_out, int out_size, void* d_ws, size_t ws_size,
                              hipStream_t stream) {
    const float* img = (const float*)d_in[0];
    float* out = (float*)d_out;
    int batch = in_sizes[0] / (IMG_H * IMG_W);   // 128
    dim3 grid(IMG_W / TILE, IMG_H / TILE, batch);
    AdaptiveThresholding_25125558681616_kernel<<<grid, NTHREADS, 0, stream>>>(img, out);
}